// CRW_65000035057877
// MI455X (gfx1250) — compile-verified
//
#include <hip/hip_runtime.h>
#include <hip/hip_bf16.h>
#include <math.h>

// Problem constants (from reference setup)
constexpr int Bb = 2, Tt = 4, Hh = 256, Ww = 256;
constexpr int NH = 7, Nn = 49, CENC = 512, Dd = 128, SPn = 64;
constexpr int Tm1 = Tt - 1;
constexpr int KP = 52;             // Nn padded up to multiple of 4 (WMMA K)
constexpr int MROWS = KP * Tt;     // 208 rows per b in padded q (13 tiles)
#define EPSL 1e-20f
#define INV_TEMP (1.0f / 0.07f)

typedef float v2f __attribute__((ext_vector_type(2)));
typedef float v8f __attribute__((ext_vector_type(8)));

// fp32 WMMA: D = A(16x4) * B(4x16) + C(16x16), wave32
__device__ __forceinline__ v8f wmma4(v2f a, v2f b, v8f c) {
  return __builtin_amdgcn_wmma_f32_16x16x4_f32(false, a, false, b, (short)0, c,
                                               false, false);
}

// ---------------------------------------------------------------------------
// Phase A: per-(b,t) 32x32-cell histograms -> window weights wts[bt][n][sp]
// sm is constant per cell: cov(cell)=1 at edges (cell 0 or 7), else 2.
// window (wi,wj) = 2x2 block of cells. wts padded to KP rows (pad rows = 0).
// ---------------------------------------------------------------------------
__global__ void k_wts(const int* __restrict__ sp_mask, float* __restrict__ wts) {
  __shared__ unsigned int hist[8 * 8 * SPn]; // [cy][cx][sp], 16 KB
  __shared__ float denomS[SPn];
  const int bt = blockIdx.x;   // b*T + t
  const int tid = threadIdx.x; // 256 threads
  for (int i = tid; i < 8 * 8 * SPn; i += 256) hist[i] = 0u;
  __syncthreads();

  const int* m = sp_mask + (size_t)bt * Hh * Ww;
  const int y = tid; // one image row per thread
  const int cy = y >> 5;
  for (int x = 0; x < Ww; ++x) {
    int sp = m[y * Ww + x] & 63;
    atomicAdd(&hist[((cy << 3) + (x >> 5)) * SPn + sp], 1u);
  }
  __syncthreads();

  if (tid < SPn) {
    unsigned int s = 0;
    for (int c = 0; c < 64; ++c) s += hist[c * SPn + tid];
    denomS[tid] = (float)s + EPSL;
  }
  __syncthreads();

  for (int idx = tid; idx < KP * SPn; idx += 256) {
    int n = idx >> 6, sp = idx & 63;
    float val = 0.f;
    if (n < Nn) {
      int wi = n / NH, wj = n % NH;
      float s = 0.f;
      for (int di = 0; di < 2; ++di)
        for (int dj = 0; dj < 2; ++dj) {
          int ci = wi + di, cj = wj + dj;
          float scale = (((ci == 0) | (ci == 7)) ? 1.f : 2.f) *
                        (((cj == 0) | (cj == 7)) ? 1.f : 2.f);
          s += (float)hist[((ci << 3) + cj) * SPn + sp] / scale;
        }
      val = s / denomS[sp];
    }
    wts[((size_t)bt * KP + n) * SPn + sp] = val;
  }
}

// ---------------------------------------------------------------------------
// Phase B: q = feats('bnct') x W_head('cd')  as padded (416 x 512)x(512 x 128)
// q layout: [b][n<KP][t][d]  (linear row l = n*T+t in [0,208) per b).
// M' = B*KP*T = 416 = 26 tiles; KP*T = 208 is 16-aligned -> b = tile/13.
// Pad rows (l >= 196): A-address clamped, store value selected to 0 (this
// also zero-fills q's K-pad rows for Phase C). No divisions in epilogue.
// grid 26 blocks x 256 thr (8 waves = 8 d-tiles); row-tile = blockIdx.
// ---------------------------------------------------------------------------
__global__ void k_gemmq(const float* __restrict__ feats,
                        const float* __restrict__ Wh, float* __restrict__ q) {
  const int wave = threadIdx.x >> 5; // col tile 0..7
  const int lane = threadIdx.x & 31;
  const int lo = lane & 15, hi = lane >> 4;
  const int b = blockIdx.x / 13;       // scalar per block (tile-aligned)
  const int rtl = blockIdx.x % 13;     // local row-tile within b
  const int l = rtl * 16 + lo;         // local A row, 0..207
  const int n = min(l >> 2, Nn - 1);   // clamp pad rows (outputs zeroed below)
  const int t = l & 3;
  // feats element (b,n,c,t) = feats[(b*Nn+n)*CENC*T + c*T + t]
  const float* arow = feats + ((size_t)(b * Nn + n) * CENC) * Tt + t;
  const int dcol = wave * 16 + lo;
  v8f c = {};
  for (int kk = 0; kk < CENC / 4; ++kk) {
    int k = kk * 4 + 2 * hi;
    v2f a, bb;
    a.x = arow[(size_t)k * Tt];
    a.y = arow[(size_t)(k + 1) * Tt];
    bb.x = Wh[(size_t)k * Dd + dcol];
    bb.y = Wh[(size_t)(k + 1) * Dd + dcol];
    c = wmma4(a, bb, c);
  }
  float* qb = q + (size_t)b * MROWS * Dd;
  const int l0 = rtl * 16 + 8 * hi; // local row of C VGPR 0 for this lane
  for (int v = 0; v < 8; ++v) {
    int lv = l0 + v;
    float val = (lv < Nn * Tt) ? c[v] : 0.f; // zero the pad rows
    qb[(size_t)lv * Dd + dcol] = val;        // always in-bounds, no branch
  }
}

// Phase B2: L2-normalize each 128-wide row of q (incl. pad rows: 0 -> 0).
// grid 416, block 128.
__global__ void k_norm(float* __restrict__ q) {
  __shared__ float red[128];
  float* row = q + (size_t)blockIdx.x * Dd;
  float v = row[threadIdx.x];
  red[threadIdx.x] = v * v;
  __syncthreads();
  for (int s = 64; s > 0; s >>= 1) {
    if (threadIdx.x < s) red[threadIdx.x] += red[threadIdx.x + s];
    __syncthreads();
  }
  float nrm = fmaxf(sqrtf(red[0]), 1e-12f);
  row[threadIdx.x] = v / nrm;
}

// ---------------------------------------------------------------------------
// Phase C: q_sp[bt][s][d] = sum_n wts[bt][n][s] * q[b][n][t][d]
// 64x128 output, K = KP = 52 (pad rows zero -> loads unconditional).
// q K-stride is T*D = 512 floats. grid 8, block 512 (16 waves).
// ---------------------------------------------------------------------------
__global__ void k_qsp(const float* __restrict__ wts, const float* __restrict__ q,
                      float* __restrict__ qsp) {
  const int bt = blockIdx.x;
  const int b = bt >> 2, t = bt & 3;
  const int wave = threadIdx.x >> 5;
  const int lane = threadIdx.x & 31;
  const int lo = lane & 15, hi = lane >> 4;
  const float* Wb = wts + (size_t)bt * KP * SPn;          // [n][s]
  const float* Qb = q + ((size_t)b * MROWS + t) * Dd;     // [n][.][d], stride 512
  float* Ob = qsp + (size_t)bt * SPn * Dd;                // [s][d]
  const int cj = wave & 7;
  const int dcol = cj * 16 + lo;
  for (int rr = 0; rr < 2; ++rr) {
    int ri = (wave >> 3) + 2 * rr; // 0..3 (s tiles)
    int srow = ri * 16 + lo;
    v8f c = {};
    for (int kk = 0; kk < KP / 4; ++kk) {
      int k = kk * 4 + 2 * hi;
      v2f a, bb;
      a.x = Wb[(size_t)k * SPn + srow];
      a.y = Wb[(size_t)(k + 1) * SPn + srow];
      bb.x = Qb[(size_t)k * (Tt * Dd) + dcol];
      bb.y = Qb[(size_t)(k + 1) * (Tt * Dd) + dcol];
      c = wmma4(a, bb, c);
    }
    for (int v = 0; v < 8; ++v)
      Ob[(size_t)(ri * 16 + v + 8 * hi) * Dd + dcol] = c[v];
  }
}

// ---------------------------------------------------------------------------
// Phase D: As[b,t,n,m] = sum_d qsp[b,t,n,d]*qsp[b,t+1,m,d]
// grid 6 (b*(T-1)+t), block 512 (16 waves -> 4x4 16x16 tiles), K=128.
// ---------------------------------------------------------------------------
__global__ void k_as(const float* __restrict__ qsp, float* __restrict__ As) {
  const int bt = blockIdx.x;
  const int b = bt / Tm1, t = bt % Tm1;
  const float* Q0 = qsp + (size_t)(b * Tt + t) * SPn * Dd;
  const float* Q1 = qsp + (size_t)(b * Tt + t + 1) * SPn * Dd;
  float* Ob = As + (size_t)bt * SPn * SPn;
  const int wave = threadIdx.x >> 5;
  const int lane = threadIdx.x & 31;
  const int lo = lane & 15, hi = lane >> 4;
  const int ri = wave >> 2, cj = wave & 3;
  const int nrow = ri * 16 + lo;
  const int mcol = cj * 16 + lo;
  v8f c = {};
  for (int kk = 0; kk < Dd / 4; ++kk) {
    int k = kk * 4 + 2 * hi;
    v2f a, bb;
    a.x = Q0[(size_t)nrow * Dd + k];
    a.y = Q0[(size_t)nrow * Dd + k + 1];
    bb.x = Q1[(size_t)mcol * Dd + k];
    bb.y = Q1[(size_t)mcol * Dd + k + 1];
    c = wmma4(a, bb, c);
  }
  for (int v = 0; v < 8; ++v)
    Ob[(size_t)(ri * 16 + v + 8 * hi) * SPn + mcol] = c[v];
}

// ---------------------------------------------------------------------------
// Phase E: zero_softmax rows of As (->A12) and of As^T (->A21).
// grid 6, block 64 (one row per thread).
// ---------------------------------------------------------------------------
__global__ void k_zsoft(const float* __restrict__ As, float* __restrict__ A12,
                        float* __restrict__ A21) {
  const int bt = blockIdx.x;
  const float* S = As + (size_t)bt * SPn * SPn;
  float* O12 = A12 + (size_t)bt * SPn * SPn;
  float* O21 = A21 + (size_t)bt * SPn * SPn;
  const int r = threadIdx.x;
  float s12 = 0.f, s21 = 0.f;
  for (int m = 0; m < SPn; ++m) {
    float e1 = expf(S[r * SPn + m] * INV_TEMP) - 1.f;
    float e2 = expf(S[m * SPn + r] * INV_TEMP) - 1.f;
    s12 += e1 * e1;
    s21 += e2 * e2;
  }
  s12 += 1e-5f;
  s21 += 1e-5f;
  for (int m = 0; m < SPn; ++m) {
    float e1 = expf(S[r * SPn + m] * INV_TEMP) - 1.f;
    float e2 = expf(S[m * SPn + r] * INV_TEMP) - 1.f;
    O12[r * SPn + m] = e1 * e1 / s12;
    O21[r * SPn + m] = e2 * e2 / s21;
  }
}

// ---------------------------------------------------------------------------
// Phase F: batched 64x64x64 fp32 matmul C[b] = A[b] @ B[b]
// grid 2 (b), block 512 (16 waves -> 4x4 tiles), K=64.
// ---------------------------------------------------------------------------
__global__ void k_mm64(const float* __restrict__ A, int sA,
                       const float* __restrict__ Bm, int sB,
                       float* __restrict__ C, int sC) {
  const int b = blockIdx.x;
  const float* Ab = A + (size_t)b * sA;
  const float* Bp = Bm + (size_t)b * sB;
  float* Cb = C + (size_t)b * sC;
  const int wave = threadIdx.x >> 5;
  const int lane = threadIdx.x & 31;
  const int lo = lane & 15, hi = lane >> 4;
  const int ri = wave >> 2, cj = wave & 3;
  const int row = ri * 16 + lo;
  const int col = cj * 16 + lo;
  v8f c = {};
  for (int kk = 0; kk < 16; ++kk) {
    int k = kk * 4 + 2 * hi;
    v2f a, b2;
    a.x = Ab[(size_t)row * 64 + k];
    a.y = Ab[(size_t)row * 64 + k + 1];
    b2.x = Bp[(size_t)k * 64 + col];
    b2.y = Bp[(size_t)(k + 1) * 64 + col];
    c = wmma4(a, b2, c);
  }
  for (int v = 0; v < 8; ++v)
    Cb[(size_t)(ri * 16 + v + 8 * hi) * 64 + col] = c[v];
}

// ---------------------------------------------------------------------------
// Phase G: loss = (1/128) * sum over both AAs and all (b,s) of
//   log(sum_c(aa[b,s,c]+eps)) - log(aa[b,s,s]+eps)
// ---------------------------------------------------------------------------
__global__ void k_loss(const float* __restrict__ P1, const float* __restrict__ P2,
                       float* __restrict__ out) {
  __shared__ float red[128];
  const int r = threadIdx.x;
  const int b = r >> 6, s = r & 63;
  float acc = 0.f;
  const float* mats[2] = {P1, P2};
  for (int i = 0; i < 2; ++i) {
    const float* row = mats[i] + ((size_t)b * SPn + s) * SPn;
    float sum = 0.f;
    for (int c = 0; c < SPn; ++c) sum += row[c] + EPSL;
    acc += logf(sum) - logf(row[s] + EPSL);
  }
  red[r] = acc;
  __syncthreads();
  for (int st = 64; st > 0; st >>= 1) {
    if (r < st) red[r] += red[r + st];
    __syncthreads();
  }
  if (r == 0) out[0] = red[0] / 128.f;
}

// ---------------------------------------------------------------------------
extern "C" void kernel_launch(void* const* d_in, const int* in_sizes, int n_in,
                              void* d_out, int out_size, void* d_ws,
                              size_t ws_size, hipStream_t stream) {
  (void)in_sizes; (void)n_in; (void)out_size; (void)ws_size;
  const float* feats = (const float*)d_in[0];  // (B,N,Cenc,T)
  const float* Wh    = (const float*)d_in[1];  // (Cenc,D)
  const int*   mask  = (const int*)d_in[2];    // (B,T,H,W)
  float* out = (float*)d_out; // [loss][P1: B*64*64][P2: B*64*64]

  float* ws  = (float*)d_ws;
  float* wts = ws;                    // B*T*KP*SP  = 26624
  float* q   = wts + Bb*Tt*KP*SPn;    // B*KP*T*D   = 53248
  float* qsp = q + Bb*MROWS*Dd;       // B*T*SP*D   = 65536
  float* As  = qsp + Bb*Tt*SPn*Dd;    // B*3*64*64  = 24576
  float* A12 = As + Bb*Tm1*SPn*SPn;   // 24576
  float* A21 = A12 + Bb*Tm1*SPn*SPn;  // 24576
  float* Lb  = A21 + Bb*Tm1*SPn*SPn;  // 8192
  float* Rb  = Lb + Bb*SPn*SPn;       // 8192
  float* Mm  = Rb + Bb*SPn*SPn;       // 8192
  float* T1  = Mm + Bb*SPn*SPn;       // 8192

  k_wts  <<<Bb * Tt, 256, 0, stream>>>(mask, wts);
  k_gemmq<<<Bb * 13, 256, 0, stream>>>(feats, Wh, q);
  k_norm <<<Bb * MROWS, 128, 0, stream>>>(q);
  k_qsp  <<<Bb * Tt, 512, 0, stream>>>(wts, q, qsp);
  k_as   <<<Bb * Tm1, 512, 0, stream>>>(qsp, As);
  k_zsoft<<<Bb * Tm1, 64, 0, stream>>>(As, A12, A21);

  const int S3 = 3 * 4096, S1 = 4096;
  float* P1 = out + 1;
  float* P2 = out + 1 + Bb * 4096;
  // L = A12_0 @ A12_1 ; R = A21_1 @ A21_0
  k_mm64<<<Bb, 512, 0, stream>>>(A12,            S3, A12 + 4096,     S3, Lb, S1);
  k_mm64<<<Bb, 512, 0, stream>>>(A21 + 4096,     S3, A21,            S3, Rb, S1);
  // P1 = L @ R
  k_mm64<<<Bb, 512, 0, stream>>>(Lb,             S1, Rb,             S1, P1, S1);
  // Mm = A12_2 @ A21_2 ; P2 = (L @ Mm) @ R
  k_mm64<<<Bb, 512, 0, stream>>>(A12 + 2 * 4096, S3, A21 + 2 * 4096, S3, Mm, S1);
  k_mm64<<<Bb, 512, 0, stream>>>(Lb,             S1, Mm,             S1, T1, S1);
  k_mm64<<<Bb, 512, 0, stream>>>(T1,             S1, Rb,             S1, P2, S1);

  k_loss<<<1, 128, 0, stream>>>(P1, P2, out);
}